// GINEConv_64707977282153
// MI455X (gfx1250) — compile-verified
//
#include <hip/hip_runtime.h>

// ---------------------------------------------------------------------------
// GINEConv for MI455X (gfx1250, wave32).
//   edge_emb = edge_attr @ We^T + be          [800k,64]
//   msg      = relu(x[row] + edge_emb)
//   agg      = segment_sum(msg, col)          [50k,64]
//   h        = relu((x+agg) @ W1^T + b1)
//   out      = h @ W2^T + b2
// Memory/atomic bound (~135 MB HBM, 4.1 GFLOP) -> keep exact f32 precision and
// use V_WMMA_F32_16X16X4_F32 for every GEMM (matrix pipe, RNE, bit-faithful
// f32 accumulate). No low-precision conversion: no speedup available from it.
// ---------------------------------------------------------------------------

typedef float v2f __attribute__((ext_vector_type(2)));
typedef float v8f __attribute__((ext_vector_type(8)));

#define N_NODES 50000
#define N_EDGES 800000
#define IN_CH   64
#define OUT_CH  64
#define EDGE_CH 32

// f32 WMMA fragment layouts (ISA 7.12.2, 32-bit data):
//  A (16x4):  lanes 0-15 hold row M=lane, K = {k,k+1}; lanes 16-31 K = {k+2,k+3}
//  B (4x16):  lanes 0-15 hold col N=lane, K = {k,k+1}; lanes 16-31 K = {k+2,k+3}
//  C/D (16x16): VGPR r: lanes 0-15 -> (M=r,   N=lane)
//                        lanes16-31 -> (M=r+8, N=lane-16)

__global__ __launch_bounds__(256) void zero_kernel(float4* __restrict__ p, int n4) {
  int i = blockIdx.x * blockDim.x + threadIdx.x;
  if (i < n4) p[i] = make_float4(0.f, 0.f, 0.f, 0.f);
}

// One wave = one tile of 16 edges x 64 channels. 50000 tiles total.
__global__ __launch_bounds__(256) void edge_kernel(
    const float* __restrict__ x,  const int*   __restrict__ ei,
    const float* __restrict__ ea, const float* __restrict__ We,
    const float* __restrict__ be, float*       __restrict__ agg) {
  const int wave  = threadIdx.x >> 5;
  const int lane  = threadIdx.x & 31;
  const int tile  = blockIdx.x * 8 + wave;
  const int e0    = tile * 16;
  const int mlane = lane & 15;
  const int khalf = (lane >> 4) * 2;   // 0 for lanes 0-15, 2 for lanes 16-31

  // Warm the gather/scatter index stream while the WMMA chain executes.
  __builtin_prefetch(ei + e0, 0, 1);            // global_prefetch_b8
  __builtin_prefetch(ei + N_EDGES + e0, 0, 1);

  v8f acc[4] = {};                     // 16 edges x 64 ch, four 16x16 N-tiles
  const float* arow = ea + (size_t)(e0 + mlane) * EDGE_CH;

  // edge_emb_tile = A(16x32) x B(32x64): 8 K-steps x 4 N-tiles of f32 WMMA.
  for (int k = 0; k < EDGE_CH; k += 4) {
    v2f a = *(const v2f*)(arow + k + khalf);
#pragma unroll
    for (int nt = 0; nt < 4; ++nt) {
      const int n = nt * 16 + mlane;                 // B[k'][n] = We[n][k']
      v2f b = *(const v2f*)(We + n * EDGE_CH + k + khalf);
      acc[nt] = __builtin_amdgcn_wmma_f32_16x16x4_f32(
          false, a, false, b, (short)0, acc[nt], false, false);
    }
  }

  // Epilogue per D layout: + be[n] + x[row][n], relu, atomic scatter to agg[col].
  const int mb = e0 + ((lane >> 4) << 3);            // e0+r (lanes 0-15) / e0+8+r
#pragma unroll
  for (int nt = 0; nt < 4; ++nt) {
    const int   n  = nt * 16 + mlane;
    const float bn = be[n];
#pragma unroll
    for (int r = 0; r < 8; ++r) {
      const int e   = mb + r;
      const int src = ei[e];               // row: gather source (half-wave uniform)
      const int dst = ei[N_EDGES + e];     // col: scatter destination
      float v = acc[nt][r] + bn + x[(size_t)src * IN_CH + n];
      v = fmaxf(v, 0.0f);
      atomicAdd(agg + (size_t)dst * IN_CH + n, v);   // global_atomic_add_f32
    }
  }
}

// One wave = one tile of 16 nodes; both MLP layers fused via per-wave LDS
// staging (no h1 round trip to HBM). 3125 tiles.
#define LDS_STRIDE 68   // 64 + 4 pad: bank-conflict-free stage of the h1 tile
__global__ __launch_bounds__(256) void mlp_kernel(
    const float* __restrict__ x,  const float* __restrict__ agg,
    const float* __restrict__ W1, const float* __restrict__ b1,
    const float* __restrict__ W2, const float* __restrict__ b2,
    float* __restrict__ out) {
  __shared__ __align__(16) float lds[8][16 * LDS_STRIDE];

  const int wave  = threadIdx.x >> 5;
  const int lane  = threadIdx.x & 31;
  const int tile  = blockIdx.x * 8 + wave;
  if (tile >= N_NODES / 16) return;                  // wave-uniform exit
  const int m0    = tile * 16;
  const int mlane = lane & 15;
  const int khalf = (lane >> 4) * 2;

  // ---- GEMM1: h1 = relu((x+agg) @ W1^T + b1), K=64 -> 16 K-steps x 4 N-tiles
  v8f acc[4] = {};
  const float* xr = x   + (size_t)(m0 + mlane) * IN_CH;
  const float* gr = agg + (size_t)(m0 + mlane) * IN_CH;
  for (int k = 0; k < IN_CH; k += 4) {
    v2f a = *(const v2f*)(xr + k + khalf) + *(const v2f*)(gr + k + khalf);
#pragma unroll
    for (int nt = 0; nt < 4; ++nt) {
      const int n = nt * 16 + mlane;
      v2f b = *(const v2f*)(W1 + n * IN_CH + k + khalf);
      acc[nt] = __builtin_amdgcn_wmma_f32_16x16x4_f32(
          false, a, false, b, (short)0, acc[nt], false, false);
    }
  }

  // Stage h1 tile (D layout -> row-major [16][68]) into this wave's LDS slab.
  float* hbuf = &lds[wave][0];
  const int mb = (lane >> 4) << 3;
#pragma unroll
  for (int nt = 0; nt < 4; ++nt) {
    const int   n  = nt * 16 + mlane;
    const float bn = b1[n];
#pragma unroll
    for (int r = 0; r < 8; ++r)
      hbuf[(mb + r) * LDS_STRIDE + n] = fmaxf(acc[nt][r] + bn, 0.0f);
  }
  // Same-wave producer/consumer: DS ops are in-order per wave; just drain DScnt.
  asm volatile("s_wait_dscnt 0" ::: "memory");

  // ---- GEMM2: out = h1 @ W2^T + b2 (A fragments re-read from LDS)
  v8f acc2[4] = {};
  const float* h1r = hbuf + mlane * LDS_STRIDE;
  for (int k = 0; k < OUT_CH; k += 4) {
    v2f a = *(const v2f*)(h1r + k + khalf);
#pragma unroll
    for (int nt = 0; nt < 4; ++nt) {
      const int n = nt * 16 + mlane;
      v2f b = *(const v2f*)(W2 + n * OUT_CH + k + khalf);
      acc2[nt] = __builtin_amdgcn_wmma_f32_16x16x4_f32(
          false, a, false, b, (short)0, acc2[nt], false, false);
    }
  }

#pragma unroll
  for (int nt = 0; nt < 4; ++nt) {
    const int   n  = nt * 16 + mlane;
    const float bn = b2[n];
#pragma unroll
    for (int r = 0; r < 8; ++r)
      out[(size_t)(m0 + mb + r) * OUT_CH + n] = acc2[nt][r] + bn;
  }
}

extern "C" void kernel_launch(void* const* d_in, const int* in_sizes, int n_in,
                              void* d_out, int out_size, void* d_ws, size_t ws_size,
                              hipStream_t stream) {
  const float* x  = (const float*)d_in[0];
  const int*   ei = (const int*)  d_in[1];
  const float* ea = (const float*)d_in[2];
  const float* We = (const float*)d_in[3];
  const float* be = (const float*)d_in[4];
  const float* W1 = (const float*)d_in[5];
  const float* b1 = (const float*)d_in[6];
  const float* W2 = (const float*)d_in[7];
  const float* b2 = (const float*)d_in[8];
  float* agg = (float*)d_ws;            // 50000*64 f32 = 12.8 MB scratch
  float* out = (float*)d_out;

  // agg must be zeroed every call (ws poisoned once; atomics accumulate).
  zero_kernel<<<(N_NODES * IN_CH / 4 + 255) / 256, 256, 0, stream>>>(
      (float4*)agg, N_NODES * IN_CH / 4);
  // 50000 edge tiles / 8 waves per block
  edge_kernel<<<N_EDGES / 16 / 8, 256, 0, stream>>>(x, ei, ea, We, be, agg);
  // 3125 node tiles / 8 waves per block
  mlp_kernel<<<(N_NODES / 16 + 7) / 8, 256, 0, stream>>>(x, agg, W1, b1, W2, b2, out);
}